// ScatterGeneralAttention_47227460386823
// MI455X (gfx1250) — compile-verified
//
#include <hip/hip_runtime.h>
#include <math.h>

// ---------------------------------------------------------------------------
// ScatterGeneralAttention for MI455X (gfx1250, wave32).
//
// Key algebraic simplification: probs = X @ (W^T a) + (b . a), so the N x D x D
// GEMM collapses to one precomputed D-vector and 1M dot products. Problem is
// memory-bound: ~2.2 GB min traffic @ 23.3 TB/s  =>  ~95us floor.
// The dot products are computed with V_WMMA_F32_16X16X4_F32 (fp32-exact).
// Segment-max subtraction is dropped: |probs| < ~1, softmax is shift-invariant.
// ---------------------------------------------------------------------------

#define DIM 256

typedef __attribute__((ext_vector_type(2))) float v2f;
typedef __attribute__((ext_vector_type(8))) float v8f;

__device__ __forceinline__ void atomic_add_f32(float* p, float v) {
    // relaxed, device-scope -> native global_atomic_add_f32 (not a CAS loop)
    __hip_atomic_fetch_add(p, v, __ATOMIC_RELAXED, __HIP_MEMORY_SCOPE_AGENT);
}

// K0: v[j] = sum_i a[i]*W[i*D+j];  vc[D] = sum_i a[i]*b[i].  One 256-thread block.
__global__ void sga_fold_kernel(const float* __restrict__ W,
                                const float* __restrict__ b,
                                const float* __restrict__ a,
                                float* __restrict__ vc) {
    const int j = threadIdx.x;
    float vj = 0.0f, c = 0.0f;
    for (int i = 0; i < DIM; ++i) {
        const float ai = a[i];
        vj = fmaf(ai, W[i * DIM + j], vj);   // coalesced across j
        c  = fmaf(ai, b[i], c);              // redundant per-thread, trivial
    }
    vc[j] = vj;
    if (j == 0) vc[DIM] = c;
}

// zero-init helper
__global__ void sga_zero_kernel(float* __restrict__ p, long long n) {
    long long i = (long long)blockIdx.x * blockDim.x + threadIdx.x;
    if (i < n) p[i] = 0.0f;
}

// K1: probs[n] = (dot(X[n], v) + c) * D^-0.5 via V_WMMA_F32_16X16X4_F32.
// One wave handles 16 rows; 64 accumulating WMMAs sweep K=0..255.
// A layout (f32 16x4): lanes 0-15 vgpr{0,1} = K{j0,j0+1}; lanes 16-31 = K{j0+2,j0+3}.
// B = v-chunk replicated across all 16 columns (same K selection per lane-half).
// D layout: lane 0 holds rows 0-7 in acc[0..7]; lane 16 holds rows 8-15.
__global__ void sga_probs_kernel(const float* __restrict__ X,
                                 const float* __restrict__ vc,
                                 float* __restrict__ probs, int N) {
    const int lane = threadIdx.x & 31;
    const int wave = threadIdx.x >> 5;
    const int half = lane >> 4;          // 0 or 1
    const int lm   = lane & 15;
    const int n0   = (blockIdx.x * (blockDim.x >> 5) + wave) * 16;
    if (n0 >= N) return;                 // wave-uniform: EXEC stays all-ones

    const float* row  = X + (long long)(n0 + lm) * DIM;
    const float* vrow = vc;
    const int coff = 2 * half;

    v8f acc = {0.f, 0.f, 0.f, 0.f, 0.f, 0.f, 0.f, 0.f};
#pragma unroll 8
    for (int j0 = 0; j0 < DIM; j0 += 4) {
        v2f af = *(const v2f*)(row + j0 + coff);   // X tile fragment (b64 load)
        v2f bf = *(const v2f*)(vrow + j0 + coff);  // v fragment (broadcast, L0-hot)
        acc = __builtin_amdgcn_wmma_f32_16x16x4_f32(
            /*neg_a=*/false, af, /*neg_b=*/false, bf,
            /*c_mod=*/(short)0, acc, /*reuse_a=*/false, /*reuse_b=*/false);
    }

    const float c     = vc[DIM];
    const float scale = 0.0625f;         // 256^-0.5
    if (lm == 0) {                       // lanes 0 and 16 hold rows 0-7 / 8-15
        float4 lo = make_float4((acc[0] + c) * scale, (acc[1] + c) * scale,
                                (acc[2] + c) * scale, (acc[3] + c) * scale);
        float4 hi = make_float4((acc[4] + c) * scale, (acc[5] + c) * scale,
                                (acc[6] + c) * scale, (acc[7] + c) * scale);
        float* dst = probs + n0 + 8 * half;
        *(float4*)(dst)     = lo;
        *(float4*)(dst + 4) = hi;
    }
}

// K2: ex[n] = exp(probs[n]) (in place); denom[idx[n]] += ex[n].
__global__ void sga_expsum_kernel(const int* __restrict__ idx,
                                  float* __restrict__ probs,
                                  float* __restrict__ denom, int N) {
    const int n = blockIdx.x * blockDim.x + threadIdx.x;
    if (n >= N) return;
    const float e = expf(probs[n]);
    probs[n] = e;
    atomic_add_f32(denom + idx[n], e);
}

// K3: s = ex[n]/denom[idx[n]]; scores[n] = s; attn[idx[n], :] += s * X[n, :].
// One wave per row; each lane owns 8 contiguous columns -> coalesced loads
// and coalesced global_atomic_add_f32 into the (L2-resident) attn buffer.
__global__ void sga_scatter_kernel(const float* __restrict__ X,
                                   const int* __restrict__ idx,
                                   const float* __restrict__ ex,
                                   const float* __restrict__ denom,
                                   float* __restrict__ scores,
                                   float* __restrict__ attn, int N) {
    const int lane = threadIdx.x & 31;
    const int wave = threadIdx.x >> 5;
    const int n = blockIdx.x * (blockDim.x >> 5) + wave;
    if (n >= N) return;
    const int seg = idx[n];
    const float s = ex[n] / denom[seg];
    if (lane == 0) scores[n] = s;

    const float* row = X + (long long)n * DIM;
    float* dst = attn + (long long)seg * DIM;
    const int c0 = lane * 8;
    float4 x0 = *(const float4*)(row + c0);
    float4 x1 = *(const float4*)(row + c0 + 4);
    atomic_add_f32(dst + c0 + 0, s * x0.x);
    atomic_add_f32(dst + c0 + 1, s * x0.y);
    atomic_add_f32(dst + c0 + 2, s * x0.z);
    atomic_add_f32(dst + c0 + 3, s * x0.w);
    atomic_add_f32(dst + c0 + 4, s * x1.x);
    atomic_add_f32(dst + c0 + 5, s * x1.y);
    atomic_add_f32(dst + c0 + 6, s * x1.z);
    atomic_add_f32(dst + c0 + 7, s * x1.w);
}

extern "C" void kernel_launch(void* const* d_in, const int* in_sizes, int n_in,
                              void* d_out, int out_size, void* d_ws, size_t ws_size,
                              hipStream_t stream) {
    // setup_inputs order: scattered_values, indices, nr_elements, W_key, b_key, attn_w
    const float* X   = (const float*)d_in[0];
    const int*   idx = (const int*)d_in[1];
    const float* W   = (const float*)d_in[3];
    const float* b   = (const float*)d_in[4];
    const float* a   = (const float*)d_in[5];

    const int N = in_sizes[1];                 // number of scattered items
    const int E = (out_size - N) / DIM;        // segments (out = scores[N] ++ attn[E*D])

    float* scores = (float*)d_out;
    float* attn   = (float*)d_out + N;

    // workspace: vc[257 pad 512] | probs/ex[N] | denom[E]   (~4.4 MB)
    float* vc    = (float*)d_ws;
    float* probs = vc + 512;
    float* denom = probs + N;

    // K0: fold W,b,attn_w into (v, c)
    sga_fold_kernel<<<1, DIM, 0, stream>>>(W, b, a, vc);

    // zero-init accumulation buffers (d_out/d_ws are poisoned by harness)
    const long long zn = (long long)E * DIM;
    sga_zero_kernel<<<(unsigned)((zn + 255) / 256), 256, 0, stream>>>(attn, zn);
    sga_zero_kernel<<<(unsigned)((E + 255) / 256), 256, 0, stream>>>(denom, E);

    // K1: WMMA dot products -> probs  (8 waves/block, 16 rows/wave)
    sga_probs_kernel<<<(unsigned)((N + 127) / 128), 256, 0, stream>>>(X, vc, probs, N);

    // K2: exp + segment denominators
    sga_expsum_kernel<<<(unsigned)((N + 255) / 256), 256, 0, stream>>>(idx, probs, denom, N);

    // K3: normalize + weighted scatter (wave per row)
    sga_scatter_kernel<<<(unsigned)((N + 7) / 8), 256, 0, stream>>>(X, idx, probs, denom,
                                                                   scores, attn, N);
}